// Edge2NodeProp1_15152644620440
// MI455X (gfx1250) — compile-verified
//
#include <hip/hip_runtime.h>

typedef __attribute__((ext_vector_type(2))) float v2f;
typedef __attribute__((ext_vector_type(8))) float v8f;

#define EDGE_DIM 128
#define N_RADIAL 16

// ---------------------------------------------------------------------------
// Zero the node accumulator (atomics accumulate into it every call).
// ---------------------------------------------------------------------------
__global__ void zero_f4_kernel(float4* __restrict__ p, long n4) {
  long i = (long)blockIdx.x * blockDim.x + threadIdx.x;
  long stride = (long)gridDim.x * blockDim.x;
  float4 z = make_float4(0.f, 0.f, 0.f, 0.f);
  for (; i < n4; i += stride) p[i] = z;
}

// ---------------------------------------------------------------------------
// Phase 1: gated = (rbf @ W_rbf^T) * x, scatter-add into h[idx_i].
// One wave owns a 16-edge tile; WMMA f32 16x16x4, K chained x4.
// All offsets are 32-bit (max index 64M) so loads/atomics use SGPR-base +
// 32-bit VGPR offset addressing with immediate byte offsets.
// ---------------------------------------------------------------------------
__global__ __launch_bounds__(256) void edge_gate_scatter_kernel(
    const float* __restrict__ x, const float* __restrict__ rbf,
    const int* __restrict__ idx_i, const float* __restrict__ W_rbf,
    float* __restrict__ h, int n_edges)
{
  const int lane = threadIdx.x & 31;
  const int r    = lane & 15;   // row/col within tile
  const int hi   = lane >> 4;   // half-wave select
  const int wave = blockIdx.x * (blockDim.x >> 5) + (threadIdx.x >> 5);
  const int nW   = gridDim.x * (blockDim.x >> 5);
  const int nFull = n_edges >> 4;

  // B tiles are loop-invariant: B[k][n] = W_rbf[n][k], n = cb*16 + r.
  v2f Bt[8][4];
#pragma unroll
  for (int cb = 0; cb < 8; ++cb) {
    const float* brow = W_rbf + (cb * 16 + r) * N_RADIAL + 2 * hi;
#pragma unroll
    for (int k0 = 0; k0 < 4; ++k0)
      Bt[cb][k0] = *(const v2f*)(brow + 4 * k0);
  }

  for (int t = wave; t < nFull; t += nW) {
    const int e0 = t << 4;

    // A tile: a[k0].{x,y} = rbf[e0 + r][4*k0 + 2*hi + {0,1}]
    v2f a[4];
    const float* arow = rbf + (e0 + r) * N_RADIAL + 2 * hi;
#pragma unroll
    for (int k0 = 0; k0 < 4; ++k0)
      a[k0] = *(const v2f*)(arow + 4 * k0);

    // node row offsets in floats (32-bit), one base + 8 immediate-offset loads
    int hoff[8];
    const int* ibase = idx_i + e0 + 8 * hi;
#pragma unroll
    for (int v = 0; v < 8; ++v)
      hoff[v] = ibase[v] * EDGE_DIM;

#pragma unroll
    for (int cb = 0; cb < 8; ++cb) {
      v8f c = {0.f, 0.f, 0.f, 0.f, 0.f, 0.f, 0.f, 0.f};
#pragma unroll
      for (int k0 = 0; k0 < 4; ++k0)
        c = __builtin_amdgcn_wmma_f32_16x16x4_f32(
            false, a[k0], false, Bt[cb][k0], (short)0, c, false, false);

      const int col = cb * 16 + r;

      // one base + 8 imm-offset loads (clause), then compute, then atomics
      const float* xbase = x + (e0 + 8 * hi) * EDGE_DIM + col;
      float xv[8];
#pragma unroll
      for (int v = 0; v < 8; ++v)
        xv[v] = xbase[v * EDGE_DIM];

#pragma unroll
      for (int v = 0; v < 8; ++v)
        atomicAdd(h + (hoff[v] + col), c[v] * xv[v]);
    }
  }

  // Scalar tail (n_edges % 16 edges): wave 0 only, each lane owns 4 channels.
  if (wave == 0) {
    for (int e = nFull << 4; e < n_edges; ++e) {
      const int nd = idx_i[e];
      const float* re = rbf + e * N_RADIAL;
#pragma unroll
      for (int cc = 0; cc < 4; ++cc) {
        const int ch = lane * 4 + cc;
        float s = 0.f;
#pragma unroll
        for (int k = 0; k < N_RADIAL; ++k) s += re[k] * W_rbf[ch * N_RADIAL + k];
        float g = s * x[e * EDGE_DIM + ch];
        atomicAdd(h + (nd * EDGE_DIM + ch), g);
      }
    }
  }
}

// ---------------------------------------------------------------------------
// Phase 2: out = act(in @ W^T + b), W torch-layout [128,128].
// W staged in LDS (64 KB). One wave computes a full 16x128 row block:
// 8 accumulator tiles, K chained as 32x WMMA f32 16x16x4.
// ---------------------------------------------------------------------------
__global__ __launch_bounds__(256) void node_mlp_kernel(
    const float* __restrict__ in, const float* __restrict__ W,
    const float* __restrict__ bias, float* __restrict__ out,
    int n_rows, int apply_swish)
{
  __shared__ float Wlds[EDGE_DIM * EDGE_DIM];
  {
    const float4* Wg = (const float4*)W;
    float4* Wl = (float4*)Wlds;
    for (int i = threadIdx.x; i < EDGE_DIM * EDGE_DIM / 4; i += blockDim.x)
      Wl[i] = Wg[i];
  }
  __syncthreads();

  const int lane = threadIdx.x & 31;
  const int r    = lane & 15;
  const int hi   = lane >> 4;
  const int wave = blockIdx.x * (blockDim.x >> 5) + (threadIdx.x >> 5);
  const int nW   = gridDim.x * (blockDim.x >> 5);
  const int nTiles = (n_rows + 15) >> 4;

  for (int t = wave; t < nTiles; t += nW) {
    const int m0 = t << 4;
    v8f acc[8];
#pragma unroll
    for (int cb = 0; cb < 8; ++cb)
      acc[cb] = (v8f){0.f, 0.f, 0.f, 0.f, 0.f, 0.f, 0.f, 0.f};

    int mrow = m0 + r;
    if (mrow >= n_rows) mrow = n_rows - 1;  // clamped load; tail stores guarded
    const float* arow = in + mrow * EDGE_DIM + 2 * hi;

#pragma unroll 4
    for (int k0 = 0; k0 < 32; ++k0) {
      v2f a = *(const v2f*)(arow + 4 * k0);
#pragma unroll
      for (int cb = 0; cb < 8; ++cb) {
        // B[k][n] = W[n][k]; n = cb*16 + r striped across lanes
        v2f b = *(const v2f*)(&Wlds[(cb * 16 + r) * EDGE_DIM + 4 * k0 + 2 * hi]);
        acc[cb] = __builtin_amdgcn_wmma_f32_16x16x4_f32(
            false, a, false, b, (short)0, acc[cb], false, false);
      }
    }

    const int rowBase = m0 + 8 * hi;
    if (m0 + 16 <= n_rows) {
      // fast path: unguarded, one base + 8 imm-offset stores per channel block
#pragma unroll
      for (int cb = 0; cb < 8; ++cb) {
        const int col = cb * 16 + r;
        const float bv = bias ? bias[col] : 0.f;
        float* obase = out + rowBase * EDGE_DIM + col;
#pragma unroll
        for (int v = 0; v < 8; ++v) {
          float val = acc[cb][v] + bv;
          if (apply_swish) val = val * (1.f / (1.f + __expf(-val)));
          obase[v * EDGE_DIM] = val;
        }
      }
    } else {
#pragma unroll
      for (int cb = 0; cb < 8; ++cb) {
        const int col = cb * 16 + r;
        const float bv = bias ? bias[col] : 0.f;
#pragma unroll
        for (int v = 0; v < 8; ++v) {
          const int m = rowBase + v;
          if (m < n_rows) {
            float val = acc[cb][v] + bv;
            if (apply_swish) val = val * (1.f / (1.f + __expf(-val)));
            out[m * EDGE_DIM + col] = val;
          }
        }
      }
    }
  }
}

// ---------------------------------------------------------------------------
extern "C" void kernel_launch(void* const* d_in, const int* in_sizes, int n_in,
                              void* d_out, int out_size, void* d_ws, size_t ws_size,
                              hipStream_t stream) {
  const float* x     = (const float*)d_in[0];
  const float* rbf   = (const float*)d_in[1];
  const int*   idx_i = (const int*)d_in[2];
  // d_in[3] = num_nodes (device scalar) -> derive from out_size instead
  const float* W_rbf = (const float*)d_in[4];
  const float* W1    = (const float*)d_in[5];
  const float* b1    = (const float*)d_in[6];
  const float* W2    = (const float*)d_in[7];
  const float* b2    = (const float*)d_in[8];
  const float* W3    = (const float*)d_in[9];
  const float* b3    = (const float*)d_in[10];
  const float* W_out = (const float*)d_in[11];
  float* outp = (float*)d_out;

  const int n_edges = in_sizes[0] / EDGE_DIM;
  const int n_nodes = out_size / EDGE_DIM;

  float* h0 = (float*)d_ws;
  float* h1 = h0 + (size_t)n_nodes * EDGE_DIM;

  // 1) zero the accumulator
  {
    long n4 = (long)n_nodes * EDGE_DIM / 4;
    long b = (n4 + 255) / 256;
    int blocks = (int)(b > 2048 ? 2048 : (b < 1 ? 1 : b));
    zero_f4_kernel<<<blocks, 256, 0, stream>>>((float4*)h0, n4);
  }

  // 2) edge gate + scatter
  {
    int nTiles = n_edges >> 4;
    int blocks = (nTiles + 7) / 8;
    if (blocks > 2048) blocks = 2048;
    if (blocks < 1) blocks = 1;
    edge_gate_scatter_kernel<<<blocks, 256, 0, stream>>>(x, rbf, idx_i, W_rbf, h0, n_edges);
  }

  // 3) node MLP: 3x (dense + swish) + linear out
  {
    int nTiles = (n_nodes + 15) / 16;
    int blocks = (nTiles + 7) / 8;
    if (blocks > 2048) blocks = 2048;
    if (blocks < 1) blocks = 1;
    node_mlp_kernel<<<blocks, 256, 0, stream>>>(h0, W1, b1, h1, n_nodes, 1);
    node_mlp_kernel<<<blocks, 256, 0, stream>>>(h1, W2, b2, h0, n_nodes, 1);
    node_mlp_kernel<<<blocks, 256, 0, stream>>>(h0, W3, b3, h1, n_nodes, 1);
    node_mlp_kernel<<<blocks, 256, 0, stream>>>(h1, W_out, nullptr, outp, n_nodes, 0);
  }
}